// ConceptSpace_37555194036374
// MI455X (gfx1250) — compile-verified
//
#include <hip/hip_runtime.h>
#include <math.h>

#define NROWS 1024
#define DDIM  256

typedef __attribute__((ext_vector_type(2))) float v2f;
typedef __attribute__((ext_vector_type(8))) float v8f;

// ---------------------------------------------------------------------------
// Kernel 1: sigmoid on both inputs -> workspace (float4-vectorized)
// ---------------------------------------------------------------------------
__global__ void __launch_bounds__(256)
sigmoid_kernel(const float* __restrict__ x1, const float* __restrict__ x2,
               float* __restrict__ sigA, float* __restrict__ sigB) {
    const int per_mat = NROWS * DDIM / 4;               // float4 count per matrix
    int idx = blockIdx.x * blockDim.x + threadIdx.x;    // 0 .. 2*per_mat-1
    const float4* src;
    float4* dst;
    if (idx < per_mat) { src = (const float4*)x1; dst = (float4*)sigA; }
    else               { src = (const float4*)x2; dst = (float4*)sigB; idx -= per_mat; }
    float4 v = src[idx];
    float4 r;
    r.x = 1.0f / (1.0f + __expf(-v.x));
    r.y = 1.0f / (1.0f + __expf(-v.y));
    r.z = 1.0f / (1.0f + __expf(-v.z));
    r.w = 1.0f / (1.0f + __expf(-v.w));
    dst[idx] = r;
}

// ---------------------------------------------------------------------------
// Kernel 2: row sums via V_WMMA_F32_16X16X4_F32 with B = ones.
// One wave32 per 16-row chunk. D = A(16x4) * ones(4x16) + C accumulates the
// row sum replicated across all 16 columns of the C tile.
// A layout (32-bit 16x4): lanes 0-15 hold M=lane, VGPR0=K0,VGPR1=K1;
//                         lanes 16-31 hold M=lane-16, VGPR0=K2,VGPR1=K3.
// C layout: VGPR r -> M=r (lanes 0-15, N=lane) / M=8+r (lanes 16-31).
// ---------------------------------------------------------------------------
__global__ void __launch_bounds__(32)
rowsum_wmma_kernel(const float* __restrict__ sig, float* __restrict__ sums) {
    const float* base = sig + (size_t)blockIdx.y * (NROWS * DDIM);
    float* s = sums + blockIdx.y * NROWS;
    const int lane  = threadIdx.x;                 // 0..31
    const int row   = blockIdx.x * 16 + (lane & 15);
    const int khalf = (lane >> 4) * 2;             // 0 or 2

    v8f c = {};
    v2f b; b.x = 1.0f; b.y = 1.0f;                 // ones matrix: layout irrelevant

    const float* rp = base + row * DDIM + khalf;
    for (int k0 = 0; k0 < DDIM; k0 += 4) {
        v2f a;
        a.x = rp[k0];
        a.y = rp[k0 + 1];
        c = __builtin_amdgcn_wmma_f32_16x16x4_f32(
                /*neg_a=*/false, a, /*neg_b=*/false, b,
                /*c_mod=*/(short)0, c, /*reuse_a=*/false, /*reuse_b=*/false);
    }
    // Column N=0 lives in lane 0 (M=0..7) and lane 16 (M=8..15).
    if (lane == 0) {
        #pragma unroll
        for (int r = 0; r < 8; ++r) s[blockIdx.x * 16 + r] = c[r];
    }
    if (lane == 16) {
        #pragma unroll
        for (int r = 0; r < 8; ++r) s[blockIdx.x * 16 + 8 + r] = c[r];
    }
}

// ---------------------------------------------------------------------------
// Kernel 3: tiled pairwise min-sum + Jaccard epilogue.
// 64x64 output tile per 256-thread block, K chunks of 64.
// LDS tiles are K-MAJOR (T[k][i]) so the inner loop uses only ds_load_b128:
//   av: broadcast per half-wave (2 addresses/wave), bv: 16 distinct 16B addrs.
// ---------------------------------------------------------------------------
#define LDS_STRIDE 68   // 64 + 4 pad floats; 272B row stride keeps 16B alignment
__global__ void __launch_bounds__(256)
jaccard_kernel(const float* __restrict__ sigA, const float* __restrict__ sigB,
               const float* __restrict__ sumA, const float* __restrict__ sumB,
               float* __restrict__ out) {
    __shared__ float As[64 * LDS_STRIDE];   // As[k][i]
    __shared__ float Bs[64 * LDS_STRIDE];   // Bs[k][j]

    const int tid = threadIdx.x;
    const int tx  = tid & 15;          // j-tile index (4 cols each)
    const int ty  = tid >> 4;          // i-tile index (4 rows each)
    const int i0  = blockIdx.y * 64;
    const int j0  = blockIdx.x * 64;

    const int lrow = tid >> 4;         // staging: row within a 16-row pass
    const int lcol = (tid & 15) * 4;   // staging: k-offset (float4 group)

    float acc[4][4] = {};

    for (int k0 = 0; k0 < DDIM; k0 += 64) {
        // Stage A and B chunks, transposed to k-major in LDS.
        #pragma unroll
        for (int p = 0; p < 4; ++p) {
            const int r = lrow + p * 16;   // 0..63
            float4 va = *(const float4*)&sigA[(i0 + r) * DDIM + k0 + lcol];
            As[(lcol + 0) * LDS_STRIDE + r] = va.x;
            As[(lcol + 1) * LDS_STRIDE + r] = va.y;
            As[(lcol + 2) * LDS_STRIDE + r] = va.z;
            As[(lcol + 3) * LDS_STRIDE + r] = va.w;
            float4 vb = *(const float4*)&sigB[(j0 + r) * DDIM + k0 + lcol];
            Bs[(lcol + 0) * LDS_STRIDE + r] = vb.x;
            Bs[(lcol + 1) * LDS_STRIDE + r] = vb.y;
            Bs[(lcol + 2) * LDS_STRIDE + r] = vb.z;
            Bs[(lcol + 3) * LDS_STRIDE + r] = vb.w;
        }
        __syncthreads();

        #pragma unroll 8
        for (int kk = 0; kk < 64; ++kk) {
            float4 av = *(const float4*)&As[kk * LDS_STRIDE + ty * 4];
            float4 bv = *(const float4*)&Bs[kk * LDS_STRIDE + tx * 4];
            const float a4[4] = {av.x, av.y, av.z, av.w};
            const float b4[4] = {bv.x, bv.y, bv.z, bv.w};
            #pragma unroll
            for (int r = 0; r < 4; ++r)
                #pragma unroll
                for (int c = 0; c < 4; ++c)
                    acc[r][c] += fminf(a4[r], b4[c]);
        }
        __syncthreads();
    }

    // Epilogue: sim = inter / (sumA_i + sumB_j - inter); write sim and sim.T.
    float sa[4], sb[4];
    #pragma unroll
    for (int r = 0; r < 4; ++r) sa[r] = sumA[i0 + ty * 4 + r];
    #pragma unroll
    for (int c = 0; c < 4; ++c) sb[c] = sumB[j0 + tx * 4 + c];

    float* outT = out + (size_t)NROWS * NROWS;
    #pragma unroll
    for (int r = 0; r < 4; ++r) {
        const int i = i0 + ty * 4 + r;
        float4 row;
        float sim[4];
        #pragma unroll
        for (int c = 0; c < 4; ++c) {
            const float inter = acc[r][c];
            sim[c] = inter / (sa[r] + sb[c] - inter);
        }
        row.x = sim[0]; row.y = sim[1]; row.z = sim[2]; row.w = sim[3];
        *(float4*)&out[(size_t)i * NROWS + j0 + tx * 4] = row;   // sim, coalesced
        #pragma unroll
        for (int c = 0; c < 4; ++c)                              // sim.T
            outT[(size_t)(j0 + tx * 4 + c) * NROWS + i] = sim[c];
    }
}

// ---------------------------------------------------------------------------
extern "C" void kernel_launch(void* const* d_in, const int* in_sizes, int n_in,
                              void* d_out, int out_size, void* d_ws, size_t ws_size,
                              hipStream_t stream) {
    (void)in_sizes; (void)n_in; (void)out_size; (void)ws_size;
    const float* x1 = (const float*)d_in[0];
    const float* x2 = (const float*)d_in[1];
    float* out = (float*)d_out;

    float* sigA = (float*)d_ws;                    // 1024*256
    float* sigB = sigA + NROWS * DDIM;             // 1024*256
    float* sumA = sigB + NROWS * DDIM;             // 1024
    // sumB = sumA + 1024 (addressed via blockIdx.y inside rowsum kernel)
    float* sumB = sumA + NROWS;

    // 1) sigmoid: 2 * 1024*256 / 4 float4s / 256 threads = 512 blocks
    sigmoid_kernel<<<512, 256, 0, stream>>>(x1, x2, sigA, sigB);

    // 2) WMMA row sums: (64 chunks of 16 rows) x (2 matrices), one wave each
    rowsum_wmma_kernel<<<dim3(64, 2), 32, 0, stream>>>(sigA, sumA);

    // 3) Jaccard: 16x16 grid of 64x64 tiles, 256 threads/block (8 wave32s)
    jaccard_kernel<<<dim3(16, 16), 256, 0, stream>>>(sigA, sigB, sumA, sumB, out);
}